// SparseBackboneEPND2_Ex1_38079180046736
// MI455X (gfx1250) — compile-verified
//
#include <hip/hip_runtime.h>

#define EPSV 1e-5f

typedef float v2f __attribute__((ext_vector_type(2)));
typedef float v8f __attribute__((ext_vector_type(8)));

// ---------------- elementwise / mask kernels ----------------

__global__ __launch_bounds__(256) void k_mask_to_float(const unsigned char* __restrict__ m,
                                                       float* __restrict__ out, long n) {
  long i = (long)blockIdx.x * 256 + threadIdx.x;
  if (i < n) out[i] = m[i] ? 1.0f : 0.0f;
}

__global__ __launch_bounds__(256) void k_mul(const float* __restrict__ a, const float* __restrict__ b,
                                             float* __restrict__ o, long n) {
  long i = (long)blockIdx.x * 256 + threadIdx.x;
  if (i < n) o[i] = a[i] * b[i];
}

// max-pool 2x2x2 downsample of the occupancy mask
__global__ __launch_bounds__(256) void k_down(const float* __restrict__ in, float* __restrict__ out,
                                              int Do, long total) {
  long i = (long)blockIdx.x * 256 + threadIdx.x;
  if (i >= total) return;
  int x = (int)(i % Do); long t = i / Do;
  int y = (int)(t % Do); t /= Do;
  int z = (int)(t % Do); int b = (int)(t / Do);
  int Di = 2 * Do;
  long Di2 = (long)Di * Di;
  const float* p = in + (long)b * Di * Di2;
  float mx = 0.0f;
  for (int dz = 0; dz < 2; dz++)
    for (int dy = 0; dy < 2; dy++)
      for (int dx = 0; dx < 2; dx++)
        mx = fmaxf(mx, p[(long)(2 * z + dz) * Di2 + (long)(2 * y + dy) * Di + (2 * x + dx)]);
  out[i] = mx;
}

// ---------------- deterministic reductions (mask count) ----------------

__global__ __launch_bounds__(256) void k_reduce_partial(const float* __restrict__ x,
                                                        float* __restrict__ partial, long n) {
  __shared__ float s[256];
  float v = 0.0f;
  for (long i = (long)blockIdx.x * 256 + threadIdx.x; i < n; i += 256L * gridDim.x) v += x[i];
  s[threadIdx.x] = v;
  __syncthreads();
  for (int o = 128; o > 0; o >>= 1) {
    if ((int)threadIdx.x < o) s[threadIdx.x] += s[threadIdx.x + o];
    __syncthreads();
  }
  if (threadIdx.x == 0) partial[blockIdx.x] = s[0];
}

__global__ void k_reduce_final(const float* __restrict__ partial, int nb, float* __restrict__ out) {
  if (threadIdx.x == 0 && blockIdx.x == 0) {
    float s = 0.0f;
    for (int i = 0; i < nb; i++) s += partial[i];
    out[0] = s;
  }
}

// ---------------- enc1: conv k4 s2 pad(0,3), IC=1 (direct VALU, cheap: ~2 GFLOP) ----------------

__global__ __launch_bounds__(256) void k_conv_e1(const float* __restrict__ xs, const float* __restrict__ w,
                                                 const float* __restrict__ m1, float* __restrict__ e1) {
  __shared__ float ws[2048];  // 32 ochan x 64 taps
  for (int i = threadIdx.x; i < 2048; i += 256) ws[i] = w[i];
  __syncthreads();
  long i = (long)blockIdx.x * 256 + threadIdx.x;
  if (i >= 524288) return;  // 2 * 64^3
  int xo = (int)(i & 63), yo = (int)((i >> 6) & 63), zo = (int)((i >> 12) & 63), b = (int)(i >> 18);
  float acc[32];
#pragma unroll
  for (int o = 0; o < 32; o++) acc[o] = 0.0f;
  const float* xp = xs + (long)b * 2097152;  // 128^3
  for (int kd = 0; kd < 4; kd++) {
    int iz = 2 * zo + kd; if (iz >= 128) break;
    for (int kh = 0; kh < 4; kh++) {
      int iy = 2 * yo + kh; if (iy >= 128) break;
      for (int kw = 0; kw < 4; kw++) {
        int ix = 2 * xo + kw; if (ix >= 128) break;
        float v = xp[(long)iz * 16384 + iy * 128 + ix];
        int tap = (kd * 4 + kh) * 4 + kw;
#pragma unroll
        for (int o = 0; o < 32; o++) acc[o] += ws[o * 64 + tap] * v;
      }
    }
  }
  float m = m1[i];
  long sp = (long)zo * 4096 + yo * 64 + xo;
#pragma unroll
  for (int o = 0; o < 32; o++)
    e1[((long)b * 32 + o) * 262144 + sp] = fmaxf(acc[o] * m, 0.0f);
}

// ---------------- weight transform: (O, IC, k^3) -> (O, k^3, IC), optional 3-D flip ----------------
// Puts consecutive input channels contiguous so WMMA A fragments load as a single b64.

__global__ __launch_bounds__(256) void k_wtrans(const float* __restrict__ w, float* __restrict__ wt,
                                                int IC, int K3, long total, int flip) {
  long i = (long)blockIdx.x * 256 + threadIdx.x;
  if (i >= total) return;
  int ic = (int)(i % IC);
  long t = i / IC;
  int tap = (int)(t % K3);
  int o = (int)(t / K3);
  int taps = flip ? (K3 - 1 - tap) : tap;  // full 3-D spatial flip == reversed flat tap index
  wt[i] = w[((long)o * IC + ic) * K3 + taps];
}

// ---------------- generic implicit-GEMM conv via V_WMMA_F32_16X16X4_F32 ----------------
// One wave computes 32 out-chans x (NT*16) consecutive-x voxels (2xNT register blocking).
// Contraction: taps outer (bounds/base computed once per tap), input channels inner in chunks of 4.
// Weights pre-transposed to (O, k^3, IC) so A fragments are contiguous b64 loads.
// Fragment layouts (ISA 7.12.2, 32-bit):
//   A 16x4 : lane holds row M=lane%16; vgpr j holds K = j + (lane>=16 ? 2 : 0)
//   B 4x16 : vgpr j holds row K = j + (lane>=16 ? 2 : 0); column N = lane%16
//   C/D    : vgpr r holds M = r + (lane>=16 ? 8 : 0); column N = lane%16
// Requires IC % 4 == 0, OC % 32 == 0, Dout % (NT*16) == 0.
template <int NT>
__global__ __launch_bounds__(256) void k_conv_wmma(
    const float* __restrict__ X, const float* __restrict__ W,
    const float* __restrict__ M, float* __restrict__ Y,
    int IC, int OC, int K, int stride, int padLo, int dil,
    int Din, int Dout, int tiles) {
  int lane = threadIdx.x & 31;
  int wid = blockIdx.x * (blockDim.x >> 5) + (threadIdx.x >> 5);
  if (wid >= tiles) return;

  int XT = Dout / (NT * 16);
  int OCT = OC >> 5;
  int t = wid;
  int xt = t % XT; t /= XT;
  int y  = t % Dout; t /= Dout;
  int z  = t % Dout; t /= Dout;
  int oct = t % OCT;
  int b   = t / OCT;

  int laneM = lane & 15;
  int hi = lane >> 4;    // 0 or 1
  int hi2 = hi << 1;     // K offset: 0 or 2
  int ocBase = oct << 5;
  int xcol[NT];
#pragma unroll
  for (int j = 0; j < NT; j++) xcol[j] = xt * (NT * 16) + j * 16 + laneM;

  int K3 = K * K * K;
  long ICK3 = (long)IC * K3;
  long Din2 = (long)Din * Din;
  long DspIn = Din2 * Din;
  long Dout2 = (long)Dout * Dout;
  long DspOut = Dout2 * Dout;

  v8f c0[NT], c1[NT];
#pragma unroll
  for (int j = 0; j < NT; j++) {
    c0[j] = (v8f){0.f, 0.f, 0.f, 0.f, 0.f, 0.f, 0.f, 0.f};
    c1[j] = (v8f){0.f, 0.f, 0.f, 0.f, 0.f, 0.f, 0.f, 0.f};
  }

  const long xBatch = (long)b * IC * DspIn;
  // transposed weight layout: wt[(o * K3 + tap) * IC + ic]
  const long wRow0 = (long)(ocBase + laneM) * ICK3;       // first 16 oc rows
  const long wRow1 = wRow0 + 16L * ICK3;                  // second 16 oc rows

  for (int kd = 0; kd < K; kd++) {
    int iz; bool okz;
    {
      int v = z * stride + kd - padLo;
      if (dil == 2) { iz = v >> 1; okz = (v >= 0) && ((v & 1) == 0) && (iz < Din); }
      else          { iz = v;      okz = (v >= 0) && (v < Din); }
    }
    if (!okz) continue;  // wave-uniform (z is uniform)
    for (int kh = 0; kh < K; kh++) {
      int iy; bool oky;
      {
        int v = y * stride + kh - padLo;
        if (dil == 2) { iy = v >> 1; oky = (v >= 0) && ((v & 1) == 0) && (iy < Din); }
        else          { iy = v;      oky = (v >= 0) && (v < Din); }
      }
      if (!oky) continue;  // wave-uniform
      for (int kw = 0; kw < K; kw++) {
        int ix[NT]; bool okx[NT];  // per-lane (xcol differs across lanes)
#pragma unroll
        for (int j = 0; j < NT; j++) {
          int v = xcol[j] * stride + kw - padLo;
          if (dil == 2) { ix[j] = v >> 1; okx[j] = (v >= 0) && ((v & 1) == 0) && (ix[j] < Din); }
          else          { ix[j] = v;      okx[j] = (v >= 0) && (v < Din); }
        }
        int tap = (kd * K + kh) * K + kw;
        long wb0 = wRow0 + (long)tap * IC + hi2;
        long wb1 = wRow1 + (long)tap * IC + hi2;
        long xb[NT];
#pragma unroll
        for (int j = 0; j < NT; j++)
          xb[j] = xBatch + (long)hi2 * DspIn + (long)iz * Din2 + (long)iy * Din + ix[j];
        for (int ic = 0; ic < IC; ic += 4) {
          v2f a0 = *(const v2f*)(W + wb0);   // contiguous: 1x b64 load
          v2f a1 = *(const v2f*)(W + wb1);
          v2f bb[NT];
#pragma unroll
          for (int j = 0; j < NT; j++) {
            bb[j].x = okx[j] ? X[xb[j]] : 0.0f;
            bb[j].y = okx[j] ? X[xb[j] + DspIn] : 0.0f;
          }
#pragma unroll
          for (int j = 0; j < NT; j++) {
            c0[j] = __builtin_amdgcn_wmma_f32_16x16x4_f32(false, a0, false, bb[j], (short)0, c0[j], false, false);
            c1[j] = __builtin_amdgcn_wmma_f32_16x16x4_f32(false, a1, false, bb[j], (short)0, c1[j], false, false);
          }
          wb0 += 4; wb1 += 4;
#pragma unroll
          for (int j = 0; j < NT; j++) xb[j] += 4 * DspIn;
        }
      }
    }
  }

  int mOff = hi << 3;
#pragma unroll
  for (int j = 0; j < NT; j++) {
    long outSp = (long)z * Dout2 + (long)y * Dout + xcol[j];
    float mval = M[(long)b * DspOut + outSp];
#pragma unroll
    for (int r = 0; r < 8; r++) {
      int oc0 = ocBase + r + mOff;
      Y[((long)b * OC + oc0) * DspOut + outSp] = c0[j][r] * mval;
      Y[((long)b * OC + oc0 + 16) * DspOut + outSp] = c1[j][r] * mval;
    }
  }
}

// ---------------- masked batch-norm (deterministic two-stage) ----------------

// one block per (b, ch); tree reduction of sum / sumsq over spatial
__global__ __launch_bounds__(256) void k_bn_stats(const float* __restrict__ Y, float* __restrict__ partial,
                                                  long Dsp, int BC) {
  __shared__ float s[256], q[256];
  const float* p = Y + (long)blockIdx.x * Dsp;
  float sv = 0.0f, qv = 0.0f;
  for (long i = threadIdx.x; i < Dsp; i += 256) { float v = p[i]; sv += v; qv += v * v; }
  s[threadIdx.x] = sv; q[threadIdx.x] = qv;
  __syncthreads();
  for (int o = 128; o > 0; o >>= 1) {
    if ((int)threadIdx.x < o) { s[threadIdx.x] += s[threadIdx.x + o]; q[threadIdx.x] += q[threadIdx.x + o]; }
    __syncthreads();
  }
  if (threadIdx.x == 0) { partial[blockIdx.x] = s[0]; partial[BC + blockIdx.x] = q[0]; }
}

__global__ void k_bn_final(const float* __restrict__ partial, float* __restrict__ stats, int C, int B) {
  int ch = blockIdx.x * blockDim.x + threadIdx.x;
  if (ch < C) {
    float s = 0.0f, q = 0.0f;
    for (int b = 0; b < B; b++) { s += partial[b * C + ch]; q += partial[B * C + b * C + ch]; }
    stats[ch] = s;
    stats[256 + ch] = q;
  }
}

__global__ __launch_bounds__(256) void k_bn_apply(float* __restrict__ Y, const float* __restrict__ stats,
                                                  const float* __restrict__ nptr, const float* __restrict__ g,
                                                  const float* __restrict__ bb, const float* __restrict__ M,
                                                  int C, long Dsp, long total) {
  long i = (long)blockIdx.x * 256 + threadIdx.x;
  if (i >= total) return;
  long sp = i % Dsp;
  long t = i / Dsp;
  int ch = (int)(t % C);
  int b = (int)(t / C);
  float n = fmaxf(nptr[0], 1.0f);
  float mean = stats[ch] / n;
  float var = stats[256 + ch] / n - mean * mean;
  float m = M[(long)b * Dsp + sp];
  float v = ((Y[i] - mean) * rsqrtf(var + EPSV) * g[ch] + bb[ch]) * m;
  Y[i] = fmaxf(v, 0.0f);
}

// ---------------- channel concat ----------------

__global__ __launch_bounds__(256) void k_concat(const float* __restrict__ A, const float* __restrict__ Bs,
                                                float* __restrict__ out, int C1, int C2, long Dsp, long total) {
  long i = (long)blockIdx.x * 256 + threadIdx.x;
  if (i >= total) return;
  long sp = i % Dsp;
  long t = i / Dsp;
  int ch = (int)(t % (C1 + C2));
  int b = (int)(t / (C1 + C2));
  out[i] = (ch < C1) ? A[((long)b * C1 + ch) * Dsp + sp]
                     : Bs[((long)b * C2 + (ch - C1)) * Dsp + sp];
}

// ---------------- host orchestration ----------------

extern "C" void kernel_launch(void* const* d_in, const int* in_sizes, int n_in,
                              void* d_out, int out_size, void* d_ws, size_t ws_size,
                              hipStream_t stream) {
  (void)in_sizes; (void)n_in; (void)out_size; (void)ws_size;

  const float* x             = (const float*)d_in[0];
  const unsigned char* mask  = (const unsigned char*)d_in[1];
  const float* w_e1  = (const float*)d_in[2];
  const float* w_e2a = (const float*)d_in[3];
  const float* g_e2a = (const float*)d_in[4];
  const float* b_e2a = (const float*)d_in[5];
  const float* w_e2b = (const float*)d_in[6];
  const float* g_e2b = (const float*)d_in[7];
  const float* b_e2b = (const float*)d_in[8];
  const float* w_e2c = (const float*)d_in[9];
  const float* g_e2c = (const float*)d_in[10];
  const float* b_e2c = (const float*)d_in[11];
  const float* w_e3  = (const float*)d_in[12];
  const float* g_e3  = (const float*)d_in[13];
  const float* b_e3  = (const float*)d_in[14];
  const float* w_e4  = (const float*)d_in[15];
  const float* g_e4  = (const float*)d_in[16];
  const float* b_e4  = (const float*)d_in[17];
  const float* w_bt  = (const float*)d_in[18];
  const float* g_bt  = (const float*)d_in[19];
  const float* b_bt  = (const float*)d_in[20];
  const float* w_d4  = (const float*)d_in[21];
  const float* g_d4  = (const float*)d_in[22];
  const float* b_d4  = (const float*)d_in[23];
  const float* w_d3  = (const float*)d_in[24];
  const float* g_d3  = (const float*)d_in[25];
  const float* b_d3  = (const float*)d_in[26];

  float* out = (float*)d_out;
  float* e1 = out;                 // 2*32*64^3  = 16777216
  float* e2 = e1 + 16777216;       // 2*64*32^3  =  4194304
  float* e3 = e2 + 4194304;        // 2*128*16^3 =  1048576
  float* e4 = e3 + 1048576;        // 2*256*16^3 =  2097152
  float* d4 = e4 + 2097152;        //             2097152
  float* d3 = d4 + 2097152;        //             1048576
  float* d2 = d3 + 1048576;        //             4194304

  // workspace layout
  float* ws = (float*)d_ws;
  size_t off = 0;
  float* m0   = ws + off; off += 4194304;   // 2*128^3
  float* m1   = ws + off; off += 524288;    // 2*64^3
  float* m2   = ws + off; off += 65536;     // 2*32^3
  float* m3   = ws + off; off += 8192;      // 2*16^3
  float* xs   = ws + off; off += 4194304;
  float* tmpA = ws + off; off += 16777216;  // h1 / cat buffers
  float* tmpB = ws + off; off += 16777216;  // h2
  float* wtbuf = ws + off; off += 1769472;  // max OC*IC*K3 (w_d4 / w_bt)
  float* statsP = ws + off; off += 1024;
  float* stats  = ws + off; off += 512;
  float* nP     = ws + off; off += 256;
  float* nbuf   = ws + off; off += 4;

  auto blocks = [](long n) { return (unsigned)((n + 255) / 256); };

  // masks + counts
  k_mask_to_float<<<blocks(4194304), 256, 0, stream>>>(mask, m0, 4194304L);
  k_down<<<blocks(524288), 256, 0, stream>>>(m0, m1, 64, 524288L);
  k_down<<<blocks(65536), 256, 0, stream>>>(m1, m2, 32, 65536L);
  k_down<<<blocks(8192), 256, 0, stream>>>(m2, m3, 16, 8192L);

  k_reduce_partial<<<256, 256, 0, stream>>>(m1, nP, 524288L);
  k_reduce_final<<<1, 32, 0, stream>>>(nP, 256, nbuf + 0);
  k_reduce_partial<<<256, 256, 0, stream>>>(m2, nP, 65536L);
  k_reduce_final<<<1, 32, 0, stream>>>(nP, 256, nbuf + 1);
  k_reduce_partial<<<256, 256, 0, stream>>>(m3, nP, 8192L);
  k_reduce_final<<<1, 32, 0, stream>>>(nP, 256, nbuf + 2);

  // xs = x * m0 ; enc1 (direct, fused mask+relu)
  k_mul<<<blocks(4194304), 256, 0, stream>>>(x, m0, xs, 4194304L);
  k_conv_e1<<<blocks(524288), 256, 0, stream>>>(xs, w_e1, m1, e1);

  // conv: transpose (+ optional flip) weights into wtbuf, then implicit-GEMM WMMA conv
  auto conv = [&](const float* X, const float* W, const float* M, float* Y,
                  int IC, int OC, int K, int stride, int padLo, int dil, int Din, int Dout, int flip) {
    int K3 = K * K * K;
    long wtotal = (long)OC * IC * K3;
    k_wtrans<<<blocks(wtotal), 256, 0, stream>>>(W, wtbuf, IC, K3, wtotal, flip);
    if (Dout >= 32) {
      int tiles = 2 * (OC / 32) * Dout * Dout * (Dout / 32);
      k_conv_wmma<2><<<(unsigned)((tiles + 7) / 8), 256, 0, stream>>>(
          X, wtbuf, M, Y, IC, OC, K, stride, padLo, dil, Din, Dout, tiles);
    } else {
      int tiles = 2 * (OC / 32) * Dout * Dout * (Dout / 16);
      k_conv_wmma<1><<<(unsigned)((tiles + 7) / 8), 256, 0, stream>>>(
          X, wtbuf, M, Y, IC, OC, K, stride, padLo, dil, Din, Dout, tiles);
    }
  };
  auto bn = [&](float* Y, int C, long Dsp, const float* g, const float* bb, const float* M, float* nslot) {
    k_bn_stats<<<(unsigned)(2 * C), 256, 0, stream>>>(Y, statsP, Dsp, 2 * C);
    k_bn_final<<<1, 256, 0, stream>>>(statsP, stats, C, 2);
    long total = 2L * C * Dsp;
    k_bn_apply<<<blocks(total), 256, 0, stream>>>(Y, stats, nslot, g, bb, M, C, Dsp, total);
  };

  // enc2 (two k5 s1 convs + one k4 s2)
  conv(e1,  w_e2a, m1, tmpA, 32, 32, 5, 1, 2, 1, 64, 64, 0);  bn(tmpA, 32, 262144L, g_e2a, b_e2a, m1, nbuf + 0);
  conv(tmpA, w_e2b, m1, tmpB, 32, 32, 5, 1, 2, 1, 64, 64, 0); bn(tmpB, 32, 262144L, g_e2b, b_e2b, m1, nbuf + 0);
  conv(tmpB, w_e2c, m2, e2, 32, 64, 4, 2, 0, 1, 64, 32, 0);   bn(e2, 64, 32768L, g_e2c, b_e2c, m2, nbuf + 1);

  // enc3 / enc4 / bottleneck
  conv(e2, w_e3, m3, e3, 64, 128, 4, 2, 0, 1, 32, 16, 0);     bn(e3, 128, 4096L, g_e3, b_e3, m3, nbuf + 2);
  conv(e3, w_e4, m3, e4, 128, 256, 3, 1, 1, 1, 16, 16, 0);    bn(e4, 256, 4096L, g_e4, b_e4, m3, nbuf + 2);
  conv(e4, w_bt, m3, d4, 256, 256, 3, 1, 1, 1, 16, 16, 0);    bn(d4, 256, 4096L, g_bt, b_bt, m3, nbuf + 2);

  // dec4: tconv s1 == conv with flipped kernel, SAME pad 1
  float* cat1 = tmpA;  // (2, 512, 16^3) = 4194304
  k_concat<<<blocks(4194304), 256, 0, stream>>>(d4, e4, cat1, 256, 256, 4096L, 4194304L);
  conv(cat1, w_d4, m3, d3, 512, 128, 3, 1, 1, 1, 16, 16, 1);  bn(d3, 128, 4096L, g_d4, b_d4, m3, nbuf + 2);

  // dec3: tconv s2 == conv with flipped kernel, lhs-dilation 2, pad lo 3
  float* cat2 = tmpA;  // (2, 256, 16^3) = 2097152
  k_concat<<<blocks(2097152), 256, 0, stream>>>(d3, e3, cat2, 128, 128, 4096L, 2097152L);
  conv(cat2, w_d3, m2, d2, 256, 64, 4, 1, 3, 2, 16, 32, 1);   bn(d2, 64, 32768L, g_d3, b_d3, m2, nbuf + 1);
}